// IIR2D_28458453303519
// MI455X (gfx1250) — compile-verified
//
#include <hip/hip_runtime.h>

// CDNA5 / gfx1250: wave32, WMMA f32 16x16x4.
typedef __attribute__((ext_vector_type(2))) float v2f;
typedef __attribute__((ext_vector_type(4))) float v4f;
typedef __attribute__((ext_vector_type(8))) float v8f;

#define W    1024
#define SEGS (W / 16)
#define TILE (16 * W)   // one 16-row x 1024-col fp32 tile = 64KB

// XOR-swizzled LDS index: kills the 16-rows-@-same-column bank conflicts of
// the WMMA B-operand reads. Key algebraic identity used for vectorization:
//   (16*s + u) ^ (4*r) = 16*(s ^ (r>>2)) + (u ^ 4*(r&3))      (u < 16)
// so intra-block swizzled offsets are per-lane constants.
__device__ __forceinline__ int lidx(int r, int c) {
    return r * W + (c ^ (r << 2));
}

// Bidirectional scan building block, TWO independent 16x1024 tiles per wave
// (two interleaved WMMA dependency chains to hide XDL latency).
// y[i] = a*y[i-1] + (1-a)*x[i], a = 0.5. Per 16-sample segment:
//   Y = M @ X + a^{i+1} (x) carry,  M[i][j] = 0.5^(i-j+1) lower-tri Toeplitz,
// with M@X as 4 chained v_wmma_f32_16x16x4_f32 per tile.
template <bool FWD>
__device__ __forceinline__ void scan2(float* lds, int lane) {
    const int  r   = lane & 15;     // A row M / B,C,D column N
    const bool hi  = lane >= 16;
    const int  lo  = r & 3;         // swizzle bits [3:2]
    const int  hi2 = r >> 2;        // swizzle bits [5:4]
    const int  rb  = r * W;

    // A-operand (constant coefficients): chunk j, VGPR p holds
    // element (M=r, K = 4j + p + 2*hi).
    v2f A[4];
#pragma unroll
    for (int j = 0; j < 4; ++j) {
        const int k0 = 4 * j + (hi ? 2 : 0);
        A[j][0] = (r >= k0)     ? __builtin_ldexpf(1.0f, -(r - k0 + 1)) : 0.0f;
        A[j][1] = (r >= k0 + 1) ? __builtin_ldexpf(1.0f, -(r - k0))     : 0.0f;
    }
    // Carry coefficients a^{M+1}; C/D VGPR v holds M = v + 8*hi.
    float ck[8];
#pragma unroll
    for (int v = 0; v < 8; ++v)
        ck[v] = __builtin_ldexpf(1.0f, -(v + (hi ? 8 : 0) + 1));

    // Per-lane constant swizzled intra-block offsets.
    // B pair loads (float2): fwd cols base+4j+2hi{,+1}; bwd mirrored.
    int offB[4];
#pragma unroll
    for (int j = 0; j < 4; ++j)
        offB[j] = FWD ? (4 * (j ^ lo) + (hi ? 2 : 0))
                      : (4 * ((3 - j) ^ lo) + (hi ? 0 : 2));
    // D quad stores (float4): fwd cols base + 8hi + 4t..+3; bwd mirrored.
    int offS[2];
#pragma unroll
    for (int t = 0; t < 2; ++t)
        offS[t] = FWD ? (4 * (((hi ? 2 : 0) + t) ^ lo))
                      : (4 * ((3 - (hi ? 2 : 0) - t) ^ lo));

    float* t0 = lds;
    float* t1 = lds + TILE;

    // Mirror border: virtual y[-1] = (pass input)[1].
    float y0 = t0[lidx(r, FWD ? 1 : (W - 2))];
    float y1 = t1[lidx(r, FWD ? 1 : (W - 2))];

    for (int s = 0; s < SEGS; ++s) {
        const int S = 16 * ((FWD ? s : (SEGS - 1 - s)) ^ hi2);
        float* p0 = t0 + rb + S;
        float* p1 = t1 + rb + S;
        v8f c0 = {}, c1 = {};
#pragma unroll
        for (int j = 0; j < 4; ++j) {
            const v2f q0 = *(const v2f*)(p0 + offB[j]);
            const v2f q1 = *(const v2f*)(p1 + offB[j]);
            v2f b0, b1;
            if (FWD) { b0 = q0; b1 = q1; }
            else     { b0[0] = q0[1]; b0[1] = q0[0];
                       b1[0] = q1[1]; b1[1] = q1[0]; }
            c0 = __builtin_amdgcn_wmma_f32_16x16x4_f32(
                     false, A[j], false, b0, (short)0, c0, false, false);
            c1 = __builtin_amdgcn_wmma_f32_16x16x4_f32(
                     false, A[j], false, b1, (short)0, c1, false, false);
        }
        // Rank-1 carry: Y[M][N] += a^{M+1} * ycarry[N]
#pragma unroll
        for (int v = 0; v < 8; ++v) { c0[v] += ck[v] * y0; c1[v] += ck[v] * y1; }
        // Next carry = Y[15][N]; M=15 lives in VGPR 7 of the high half.
        const float n0 = __shfl(c0[7], r + 16, 32);
        const float n1 = __shfl(c1[7], r + 16, 32);
        // Store D as two float4 quads per tile (reversed for anticausal).
#pragma unroll
        for (int t = 0; t < 2; ++t) {
            v4f w0, w1;
#pragma unroll
            for (int u = 0; u < 4; ++u) {
                const int src = FWD ? (4 * t + u) : (4 * t + 3 - u);
                w0[u] = c0[src];
                w1[u] = c1[src];
            }
            *(v4f*)(p0 + offS[t]) = w0;
            *(v4f*)(p1 + offS[t]) = w1;
        }
        y0 = n0;
        y1 = n1;
    }
}

// Pass 1: bidirectional scan along W. One wave per pair of 16-row strips.
__global__ void __launch_bounds__(32)
iir_rowpass(const float* __restrict__ in, float* __restrict__ out) {
    extern __shared__ float lds[];
    const int lane = threadIdx.x;
    const size_t po =
        (size_t)blockIdx.y * ((size_t)W * W) + (size_t)blockIdx.x * 32 * W;

#pragma unroll
    for (int t = 0; t < 2; ++t) {
        const float* src = in + po + (size_t)(t * 16) * W;
        float* tl = lds + t * TILE;
        for (int i = lane; i < 16 * (W / 4); i += 32) {
            const int rr = i >> 8, c4 = (i & 255) << 2;
            *(v4f*)&tl[lidx(rr, c4)] = *(const v4f*)(src + rr * W + c4);
        }
    }
    __syncthreads();
    scan2<true>(lds, lane);
    __syncthreads();
    scan2<false>(lds, lane);
    __syncthreads();
#pragma unroll
    for (int t = 0; t < 2; ++t) {
        float* dst = out + po + (size_t)(t * 16) * W;
        const float* tl = lds + t * TILE;
        for (int i = lane; i < 16 * (W / 4); i += 32) {
            const int rr = i >> 8, c4 = (i & 255) << 2;
            *(v4f*)(dst + rr * W + c4) = *(const v4f*)&tl[lidx(rr, c4)];
        }
    }
}

// Pass 2: bidirectional scan along H, in place on d_out. One wave per pair
// of 16-column strips, transposed through LDS so scan2 is reused verbatim.
__global__ void __launch_bounds__(32)
iir_colpass(float* __restrict__ data) {
    extern __shared__ float lds[];
    const int lane = threadIdx.x;
    float* plane = data + (size_t)blockIdx.y * ((size_t)W * W);

#pragma unroll
    for (int t = 0; t < 2; ++t) {
        const float* base = plane + (2 * blockIdx.x + t) * 16;
        float* tl = lds + t * TILE;
        for (int i = lane; i < W * 4; i += 32) {
            const int h = i >> 2, g = (i & 3) << 2;
            const v4f v = *(const v4f*)(base + (size_t)h * W + g);
            tl[lidx(g + 0, h)] = v[0];
            tl[lidx(g + 1, h)] = v[1];
            tl[lidx(g + 2, h)] = v[2];
            tl[lidx(g + 3, h)] = v[3];
        }
    }
    __syncthreads();
    scan2<true>(lds, lane);
    __syncthreads();
    scan2<false>(lds, lane);
    __syncthreads();
#pragma unroll
    for (int t = 0; t < 2; ++t) {
        float* base = plane + (2 * blockIdx.x + t) * 16;
        const float* tl = lds + t * TILE;
        for (int i = lane; i < W * 4; i += 32) {
            const int h = i >> 2, g = (i & 3) << 2;
            v4f v;
            v[0] = tl[lidx(g + 0, h)];
            v[1] = tl[lidx(g + 1, h)];
            v[2] = tl[lidx(g + 2, h)];
            v[3] = tl[lidx(g + 3, h)];
            *(v4f*)(base + (size_t)h * W + g) = v;
        }
    }
}

extern "C" void kernel_launch(void* const* d_in, const int* in_sizes, int n_in,
                              void* d_out, int out_size, void* d_ws, size_t ws_size,
                              hipStream_t stream) {
    (void)in_sizes; (void)n_in; (void)out_size; (void)d_ws; (void)ws_size;
    const float* x   = (const float*)d_in[0];
    float*       out = (float*)d_out;

    const size_t shmem = 2 * TILE * sizeof(float);   // 128 KB (WGP has 320 KB)
    // Opt in to >64KB dynamic LDS. Not a stream op: safe under graph capture,
    // deterministic, idempotent.
    hipFuncSetAttribute((const void*)iir_rowpass,
                        hipFuncAttributeMaxDynamicSharedMemorySize, (int)shmem);
    hipFuncSetAttribute((const void*)iir_colpass,
                        hipFuncAttributeMaxDynamicSharedMemorySize, (int)shmem);

    const dim3 grid(32, 64);   // 32 strip-pairs x 64 planes
    iir_rowpass<<<grid, 32, shmem, stream>>>(x, out);
    iir_colpass<<<grid, 32, shmem, stream>>>(out);
}